// SynthLocalLoss_61340722921561
// MI455X (gfx1250) — compile-verified
//
#include <hip/hip_runtime.h>
#include <hip/hip_bf16.h>
#include <stdint.h>

#define BB 4
#define ZDIM 40
#define YDIM 1024
#define XDIM 1024
#define NR 262144
#define NL 1048576
#define TPB 256
#define NBLK 256
#define TILES 4
#define EMPTY_KEY 0xFFFFFFFFu

typedef __attribute__((ext_vector_type(2))) float v2f;
typedef __attribute__((ext_vector_type(8))) float v8f;

__device__ __forceinline__ uint32_t hslot(uint32_t cell, uint32_t shift) {
  return (cell * 2654435761u) >> shift;   // Knuth multiplicative, top bits
}

// ---------------- hash table build (L2-resident, 2^21..2^22 slots) ---------

__global__ __launch_bounds__(TPB) void ht_init(uint32_t* __restrict__ keys,
                                               int* __restrict__ vals) {
  uint32_t i = blockIdx.x * TPB + threadIdx.x;
  keys[i] = EMPTY_KEY;
  vals[i] = 0x7FFFFFFF;
}

__global__ __launch_bounds__(TPB) void ht_insert(const int4* __restrict__ li,
                                                 uint32_t* __restrict__ keys,
                                                 int* __restrict__ vals,
                                                 uint32_t mask, uint32_t shift) {
  int i = blockIdx.x * TPB + threadIdx.x;           // exact grid, no bounds check
  int4 r = li[i];                                   // b,z,y,x
  uint32_t cell = (uint32_t)(((r.x * ZDIM + r.y) * YDIM + r.z) * XDIM + r.w);
  uint32_t slot = hslot(cell, shift);
  for (;;) {
    uint32_t prev = atomicCAS(&keys[slot], EMPTY_KEY, cell);
    if (prev == EMPTY_KEY || prev == cell) {
      atomicMin(&vals[slot], i);                    // stable-argsort semantics: min row
      return;
    }
    slot = (slot + 1) & mask;
  }
}

__device__ __forceinline__ int ht_lookup(const uint32_t* __restrict__ keys,
                                         const int* __restrict__ vals,
                                         uint32_t cell, uint32_t mask,
                                         uint32_t shift) {
  uint32_t slot = hslot(cell, shift);
  for (;;) {
    uint32_t k = keys[slot];
    if (k == cell) return vals[slot];
    if (k == EMPTY_KEY) return -1;
    slot = (slot + 1) & mask;
  }
}

// ---------------- CDNA5 async global->LDS staging --------------------------

__device__ __forceinline__ uint32_t lds_off(const void* p) {
  // isLDS aperture lives in addr[63:32]; low 32 bits of a flat LDS pointer
  // are the raw LDS byte offset (ISA 10.2).
  return (uint32_t)(uintptr_t)p;
}
__device__ __forceinline__ void async_ld_b128(uint32_t lds, const void* g) {
  asm volatile("global_load_async_to_lds_b128 %0, %1, off"
               :: "v"(lds), "v"((unsigned long long)(uintptr_t)g) : "memory");
}
__device__ __forceinline__ void async_ld_b32(uint32_t lds, const void* g) {
  asm volatile("global_load_async_to_lds_b32 %0, %1, off"
               :: "v"(lds), "v"((unsigned long long)(uintptr_t)g) : "memory");
}

__device__ __forceinline__ float smoothl1(float d) {
  float ad = fabsf(d);
  return ad < 1.0f ? 0.5f * d * d : ad - 0.5f;
}

// ---------------- match + per-block loss partials --------------------------

__global__ __launch_bounds__(TPB) void match_kernel(
    const float4* __restrict__ pf4, const float* __restrict__ occ,
    const int4* __restrict__ ridx, const float* __restrict__ lfeat,
    const uint32_t* __restrict__ keys, const int* __restrict__ vals,
    uint32_t mask, uint32_t shift, float* __restrict__ partials) {
  __shared__ int4   s_idx[2][TPB];
  __shared__ float4 s_pf [2][TPB];
  __shared__ float  s_oc [2][TPB];
  __shared__ float  s_red[TPB];
  const int tid = threadIdx.x, bid = blockIdx.x;

  float a_occ = 0.f, a_cnt = 0.f, a_off = 0.f, a_feat = 0.f;

  {                                                   // prefetch tile 0
    int i = (bid * TILES + 0) * TPB + tid;
    async_ld_b128(lds_off(&s_idx[0][tid]), &ridx[i]);
    async_ld_b128(lds_off(&s_pf [0][tid]), &pf4[i]);
    async_ld_b32 (lds_off(&s_oc [0][tid]), &occ[i]);
  }

  for (int k = 0; k < TILES; ++k) {
    __builtin_amdgcn_s_wait_asynccnt(0);              // my tile-k loads landed
    __syncthreads();                                  // everyone's landed
    if (k + 1 < TILES) {                              // overlap: async(k+1) || compute(k)
      int nb = (k + 1) & 1;
      int i = (bid * TILES + k + 1) * TPB + tid;
      async_ld_b128(lds_off(&s_idx[nb][tid]), &ridx[i]);
      async_ld_b128(lds_off(&s_pf [nb][tid]), &pf4[i]);
      async_ld_b32 (lds_off(&s_oc [nb][tid]), &occ[i]);
    }
    const int buf = k & 1;
    int4 r  = s_idx[buf][tid];
    float4 p = s_pf[buf][tid];
    float  l = s_oc[buf][tid];
    const int b = r.x, z = r.y, y = r.z, x = r.w;

    int bestd = 1 << 30, bestrow = -1, bz = 0, by = 0, bx = 0;
#pragma unroll
    for (int dz = -1; dz <= 1; ++dz) {
      int czz = min(max(z + dz, 0), ZDIM - 1);
      int adz = dz < 0 ? -dz : dz;
#pragma unroll
      for (int dy = -1; dy <= 1; ++dy) {
        int cyy = min(max(y + dy, 0), YDIM - 1);
        int ady = dy < 0 ? -dy : dy;
#pragma unroll
        for (int dx = -1; dx <= 1; ++dx) {
          int cxx = min(max(x + dx, 0), XDIM - 1);
          int d = adz + ady + (dx < 0 ? -dx : dx);    // L1 of UNclipped offset
          if (d < bestd) {                            // strict < keeps earliest k on tie
            uint32_t cell = (uint32_t)(((b * ZDIM + czz) * YDIM + cyy) * XDIM + cxx);
            int row = ht_lookup(keys, vals, cell, mask, shift);
            if (row >= 0) { bestd = d; bestrow = row; bz = czz; by = cyy; bx = cxx; }
          }
        }
      }
    }

    float m  = (bestrow >= 0) ? 1.f : 0.f;
    float sp = fmaxf(l, 0.f) + log1pf(expf(-fabsf(l)));   // logaddexp(0,l)
    a_occ += sp - l * m;
    if (bestrow >= 0) {
      a_cnt += 1.f;
      float d0 = p.x - (float)(bx - x);               // gt_d reversed: x,y,z
      float d1 = p.y - (float)(by - y);
      float d2 = p.z - (float)(bz - z);
      a_off  += smoothl1(d0) + smoothl1(d1) + smoothl1(d2);
      a_feat += fabsf(p.w - lfeat[bestrow * 4 + 3]);
    }
  }

  float accs[4] = {a_occ, a_cnt, a_off, a_feat};      // fixed-order tree reduce
#pragma unroll
  for (int q = 0; q < 4; ++q) {
    __syncthreads();
    s_red[tid] = accs[q];
    __syncthreads();
    for (int s = TPB / 2; s > 0; s >>= 1) {
      if (tid < s) s_red[tid] += s_red[tid + s];
      __syncthreads();
    }
    if (tid == 0) partials[bid * 4 + q] = s_red[0];
  }
}

// ---------------- finalize: WMMA-based cross-block reduction ----------------
// Single wave (EXEC all ones as WMMA requires). A = ones(16x4); each wmma
// folds 64 block-partials into C; row 0 of D is then lane-summed. Result is
// the exact total independent of B's K-lane split because A is all-ones.

__global__ __launch_bounds__(32) void finalize_kernel(
    const float* __restrict__ partials, float* __restrict__ out) {
  const int lane = threadIdx.x;
  const int col = lane & 15, half = lane >> 4;
  v2f a; a.x = 1.f; a.y = 1.f;
  float sums[4];
#pragma unroll
  for (int q = 0; q < 4; ++q) {
    v8f c = {0.f, 0.f, 0.f, 0.f, 0.f, 0.f, 0.f, 0.f};
#pragma unroll
    for (int chunk = 0; chunk < NBLK / 64; ++chunk) {
      int idx = chunk * 64 + 4 * col + 2 * half;      // each of 64 partials once
      v2f bv;
      bv.x = partials[idx * 4 + q];
      bv.y = partials[(idx + 1) * 4 + q];
      c = __builtin_amdgcn_wmma_f32_16x16x4_f32(false, a, false, bv,
                                                (short)0, c, false, false);
    }
    float v = c[0];                                   // D[0][n], lanes 0..15
    for (int off = 8; off >= 1; off >>= 1) v += __shfl_xor(v, off, 16);
    sums[q] = v;
  }
  if (lane == 0) {
    float occ_loss  = sums[0] / (float)NR;
    float cnt       = sums[1];
    float off_loss  = sums[2] / fmaxf(cnt * 3.f, 1.f);
    float feat_loss = sums[3] / fmaxf(cnt, 1.f);
    out[0] = 0.2f * occ_loss + off_loss + feat_loss;
  }
}

// ---------------- host launch ----------------------------------------------

extern "C" void kernel_launch(void* const* d_in, const int* in_sizes, int n_in,
                              void* d_out, int out_size, void* d_ws, size_t ws_size,
                              hipStream_t stream) {
  (void)in_sizes; (void)n_in; (void)out_size;
  const float4* pf4  = (const float4*)d_in[0];   // pred_feat (Nr,4)
  const float*  occp = (const float*)d_in[1];    // pred_occ  (Nr,1,1)
  const int4*   ridx = (const int4*)d_in[2];     // radar_indices (Nr,4)
  const float*  lfe  = (const float*)d_in[3];    // lidar_features (Nl,4)
  const int4*   lidx = (const int4*)d_in[4];     // lidar_indices (Nl,4)
  // d_in[5] = R == 1 (compile-time specialized)

  uint32_t log2cap = 21;                         // 16 MB table minimum (load 0.5)
  if (ws_size >= (((size_t)8u) << 22) + 8192) log2cap = 22;  // 32 MB (load 0.25)
  uint32_t cap = 1u << log2cap;
  uint32_t mask = cap - 1, shift = 32 - log2cap;

  uint32_t* keys = (uint32_t*)d_ws;
  int* vals = (int*)(keys + cap);
  float* partials = (float*)(vals + cap);

  ht_init<<<cap / TPB, TPB, 0, stream>>>(keys, vals);
  ht_insert<<<NL / TPB, TPB, 0, stream>>>(lidx, keys, vals, mask, shift);
  match_kernel<<<NBLK, TPB, 0, stream>>>(pf4, occp, ridx, lfe, keys, vals,
                                         mask, shift, partials);
  finalize_kernel<<<1, 32, 0, stream>>>(partials, (float*)d_out);
}